// Agnn_83330955477198
// MI455X (gfx1250) — compile-verified
//
#include <hip/hip_runtime.h>
#include <hip/hip_bf16.h>
#include <math.h>

typedef __attribute__((ext_vector_type(2))) float v2f;
typedef __attribute__((ext_vector_type(8))) float v8f;

#define AGNN_EPS 1e-12f

// Float atomic max via int/uint punning (always available, correct for mixed signs
// when the cell is initialized to -inf = 0xFF800000).
__device__ __forceinline__ void atomicMaxF32(float* addr, float val) {
    if (val >= 0.0f) {
        atomicMax((int*)addr, __float_as_int(val));
    } else {
        atomicMin((unsigned int*)addr, __float_as_uint(val));
    }
}

// ---------------------------------------------------------------------------
// Kernel 1: per-node L2 normalize + init amax/denom + zero the accumulator
// (accumulator lives in d_out). One thread per node, float4 row traffic.
// ---------------------------------------------------------------------------
__global__ void agnn_node_prep(const float* __restrict__ x,
                               float* __restrict__ xn,
                               float* __restrict__ amax,
                               float* __restrict__ denom,
                               float* __restrict__ acc,
                               int N) {
    int t = blockIdx.x * blockDim.x + threadIdx.x;
    if (t >= N) return;
    const float4* xr = (const float4*)(x + (size_t)t * 32);
    float4 v[8];
    float s = 0.f;
#pragma unroll
    for (int i = 0; i < 8; ++i) {
        v[i] = xr[i];
        s += v[i].x * v[i].x + v[i].y * v[i].y + v[i].z * v[i].z + v[i].w * v[i].w;
    }
    float inv = 1.0f / fmaxf(sqrtf(s), AGNN_EPS);
    float4* xo = (float4*)(xn + (size_t)t * 32);
    float4* ao = (float4*)(acc + (size_t)t * 32);
    float4 z = make_float4(0.f, 0.f, 0.f, 0.f);
#pragma unroll
    for (int i = 0; i < 8; ++i) {
        float4 o = v[i];
        o.x *= inv; o.y *= inv; o.z *= inv; o.w *= inv;
        xo[i] = o;
        ao[i] = z;
    }
    amax[t]  = __int_as_float(0xFF800000); // -inf
    denom[t] = 0.f;
}

// ---------------------------------------------------------------------------
// Kernel 2: per-edge attention logits via WMMA.
// One wave handles 16 edges. A = xn[dst] rows, B = xn[src] cols; 8 chained
// v_wmma_f32_16x16x4_f32 cover K=32; diagonal of the 16x16 accumulator is the
// per-edge dot product. Lane l supplies A/B elements (row l%16,
// K = 4*chunk + 2*(l>>4) + {0,1}) per the CDNA5 f32 A/B VGPR layouts.
// Control flow is wave-uniform around the WMMAs (EXEC all-ones requirement).
// ---------------------------------------------------------------------------
__global__ void agnn_edge_alpha(const float* __restrict__ xn,
                                const int* __restrict__ ei,   // [2*E] src row then dst row
                                const float* __restrict__ beta,
                                float* __restrict__ alpha,    // [ET]
                                float* __restrict__ amax,     // [N]
                                int E, int ET) {
    int wave = (blockIdx.x * blockDim.x + threadIdx.x) >> 5;
    int lane = threadIdx.x & 31;
    long long base = (long long)wave * 16;
    if (base >= (long long)ET) return;       // uniform per wave

    int m  = lane & 15;                       // edge within group
    int kh = lane >> 4;                       // K-half select
    long long eReq = base + m;
    long long e = (eReq < (long long)ET) ? eReq : (long long)(ET - 1); // clamp tail
    bool real = e < (long long)E;
    int sl = (int)(e - (long long)E);         // self-loop node id when !real
    int s = real ? ei[e] : sl;                // source j
    int d = real ? ei[(long long)E + e] : sl; // target i

    const float* ra = xn + (size_t)d * 32;    // A row (dst, normalized)
    const float* rb = xn + (size_t)s * 32;    // B col (src, normalized)

    v8f c = {};
#pragma unroll
    for (int ck = 0; ck < 8; ++ck) {
        int cb = ck * 4 + kh * 2;
        v2f a = { ra[cb], ra[cb + 1] };
        v2f b = { rb[cb], rb[cb + 1] };
        // (neg_a, A, neg_b, B, c_mod, C, reuse_a, reuse_b)
        c = __builtin_amdgcn_wmma_f32_16x16x4_f32(
                false, a, false, b, (short)0, c, false, false);
    }

    // Diagonal D[m][m]: m<8 -> lane m, c[m]; m>=8 -> lane m+16, c[m-8].
    bool diagLane = (lane < 8) || (lane >= 24);
    if (diagLane && eReq < (long long)ET) {
        int ci = (lane < 8) ? lane : (lane - 24);
        float av = beta[0] * c[ci];
        alpha[eReq] = av;
        atomicMaxF32(amax + d, av);
    }
}

// ---------------------------------------------------------------------------
// Kernel 3: softmax numerator + weighted scatter. One wave per edge:
// 32 lanes cover the 32 features -> coalesced 128B row gather of x[src] and
// coalesced 128B of global_atomic_add_f32 into the accumulator row of dst.
// ---------------------------------------------------------------------------
__global__ void agnn_edge_accum(const float* __restrict__ x,
                                const int* __restrict__ ei,
                                const float* __restrict__ alpha,
                                const float* __restrict__ amax,
                                float* __restrict__ denom,
                                float* __restrict__ acc,
                                int E, int ET) {
    long long wave = (long long)blockIdx.x * (blockDim.x >> 5) + (threadIdx.x >> 5);
    int lane = threadIdx.x & 31;
    if (wave >= (long long)ET) return;
    long long e = wave;
    bool real = e < (long long)E;
    int sl = (int)(e - (long long)E);
    int s = real ? ei[e] : sl;
    int d = real ? ei[(long long)E + e] : sl;

    float ex = __expf(alpha[e] - amax[d]);
    if (lane == 0) atomicAdd(denom + d, ex);
    float xv = x[(size_t)s * 32 + lane];
    atomicAdd(acc + (size_t)d * 32 + lane, ex * xv);
}

// ---------------------------------------------------------------------------
// Kernel 4: out = x + relu(acc / max(denom, eps)), in place over d_out.
// ---------------------------------------------------------------------------
__global__ void agnn_finalize(const float* __restrict__ x,
                              const float* __restrict__ denom,
                              float* __restrict__ out,
                              int total) {
    int i = blockIdx.x * blockDim.x + threadIdx.x;
    if (i >= total) return;
    float v = out[i] / fmaxf(denom[i >> 5], AGNN_EPS);
    out[i] = x[i] + fmaxf(v, 0.f);
}

extern "C" void kernel_launch(void* const* d_in, const int* in_sizes, int n_in,
                              void* d_out, int out_size, void* d_ws, size_t ws_size,
                              hipStream_t stream) {
    const float* x    = (const float*)d_in[0]; // [N,32] f32
    const int*   ei   = (const int*)d_in[1];   // [2,E] integer
    const float* beta = (const float*)d_in[2]; // [1] f32

    const int D  = 32;
    const int N  = in_sizes[0] / D;
    const int E  = in_sizes[1] / 2;
    const int ET = E + N;                       // edges + self-loops

    float* out = (float*)d_out;                 // doubles as the accumulator

    // Workspace layout (all 16B-aligned given hipMalloc alignment of d_ws):
    //   xn    : N*32 floats  (6.4 MB)
    //   amax  : N floats
    //   denom : N floats
    //   alpha : ET floats    (6.6 MB)
    float* xn    = (float*)d_ws;
    float* amax  = xn + (size_t)N * D;
    float* denom = amax + N;
    float* alpha = denom + N;

    // 1) normalize + init
    {
        int blocks = (N + 255) / 256;
        agnn_node_prep<<<blocks, 256, 0, stream>>>(x, xn, amax, denom, out, N);
    }
    // 2) edge logits via WMMA + atomic max
    {
        long long waves   = ((long long)ET + 15) / 16;
        long long threads = waves * 32;
        int blocks = (int)((threads + 255) / 256);
        agnn_edge_alpha<<<blocks, 256, 0, stream>>>(xn, ei, beta, alpha, amax, E, ET);
    }
    // 3) exp + denom + weighted scatter (one wave per edge)
    {
        long long threads = (long long)ET * 32;
        int blocks = (int)((threads + 255) / 256);
        agnn_edge_accum<<<blocks, 256, 0, stream>>>(x, ei, alpha, amax, denom, out, E, ET);
    }
    // 4) finalize
    {
        int total = N * D;
        int blocks = (total + 255) / 256;
        agnn_finalize<<<blocks, 256, 0, stream>>>(x, denom, out, total);
    }
}